// AELoss_22316650070104
// MI455X (gfx1250) — compile-verified
//
#include <hip/hip_runtime.h>

// Associative-embedding push/pull loss, MI455X (gfx1250, wave32).
// Phase 1: bandwidth-bound segmented reduction (count/sum/sumsq) via LDS f32 atomics.
// Phase 2: per-row finalize; push term's cross-product m_i*m_j computed with
//          V_WMMA_F32_16X16X4_F32 rank-1 outer products over 9x9 16x16 tiles.

#define K_SEG   129
#define K_PAD   144          // 9 tiles of 16
#define ROW_N   131072
#define CHUNKS  64           // chunks per row -> 8192 blocks total
#define BDIM    256

typedef __attribute__((ext_vector_type(2))) float v2f;
typedef __attribute__((ext_vector_type(8))) float v8f;

__global__ void zero_ws_kernel(float* __restrict__ ws, int n) {
    int i = blockIdx.x * blockDim.x + threadIdx.x;
    if (i < n) ws[i] = 0.0f;
}

__global__ void seg_reduce_kernel(const float* __restrict__ tags,
                                  const int*   __restrict__ gt,
                                  float*       __restrict__ ws) {
    __shared__ float s_cnt[K_SEG];
    __shared__ float s_sum[K_SEG];
    __shared__ float s_sq [K_SEG];

    const int tid = threadIdx.x;
    if (tid < K_SEG) { s_cnt[tid] = 0.0f; s_sum[tid] = 0.0f; s_sq[tid] = 0.0f; }
    __syncthreads();

    const int b = blockIdx.y;
    const long long row = (long long)b * ROW_N;
    const int chunk_elems = ROW_N / CHUNKS;            // 2048
    const int chunk_base  = blockIdx.x * chunk_elems;

    const float4* t4 = reinterpret_cast<const float4*>(tags + row + chunk_base);
    const int4*   g4 = reinterpret_cast<const int4*  >(gt   + row + chunk_base);
    const int nvec = chunk_elems / 4;                  // 512

    for (int v = tid; v < nvec; v += BDIM) {
        float4 x = t4[v];
        int4   g = g4[v];
        atomicAdd(&s_cnt[g.x], 1.0f); atomicAdd(&s_sum[g.x], x.x); atomicAdd(&s_sq[g.x], x.x * x.x);
        atomicAdd(&s_cnt[g.y], 1.0f); atomicAdd(&s_sum[g.y], x.y); atomicAdd(&s_sq[g.y], x.y * x.y);
        atomicAdd(&s_cnt[g.z], 1.0f); atomicAdd(&s_sum[g.z], x.z); atomicAdd(&s_sq[g.z], x.z * x.z);
        atomicAdd(&s_cnt[g.w], 1.0f); atomicAdd(&s_sum[g.w], x.w); atomicAdd(&s_sq[g.w], x.w * x.w);
    }
    __syncthreads();

    if (tid < K_SEG) {
        float* wrow = ws + (size_t)b * (3 * K_PAD);
        atomicAdd(&wrow[tid],             s_cnt[tid]);
        atomicAdd(&wrow[K_PAD + tid],     s_sum[tid]);
        atomicAdd(&wrow[2 * K_PAD + tid], s_sq [tid]);
    }
}

// One 32-thread wave per batch row.
__global__ void finalize_kernel(const float* __restrict__ ws,
                                float* __restrict__ out, int B) {
    __shared__ float s_mean [K_PAD];
    __shared__ float s_valid[K_PAD];

    const int lane = threadIdx.x;      // 0..31, wave32
    const int b    = blockIdx.x;
    const float* wrow = ws + (size_t)b * (3 * K_PAD);

    float pull_acc = 0.0f, ntag_acc = 0.0f;
    for (int k = lane; k < K_PAD; k += 32) {
        float mean = 0.0f, vld = 0.0f;
        if (k < K_SEG) {
            float c = wrow[k];
            if (c > 0.0f) {
                float s = wrow[K_PAD + k];
                float q = wrow[2 * K_PAD + k];
                mean = s / c;
                vld  = 1.0f;
                float sq = q - c * mean * mean;   // sum over group of (x-mean)^2
                pull_acc += sq / c;
                ntag_acc += 1.0f;
            }
        }
        s_mean[k]  = mean;
        s_valid[k] = vld;
    }
    for (int m = 16; m >= 1; m >>= 1) {
        pull_acc += __shfl_xor(pull_acc, m, 32);
        ntag_acc += __shfl_xor(ntag_acc, m, 32);
    }
    __syncthreads();

    // push: sum over valid pairs of exp(-(m_i - m_j)^2).
    // cross term m_i*m_j via WMMA f32 16x16x4 rank-1 outer product per tile pair.
    float acc  = 0.0f;
    const int half = lane >> 4;        // 0: lanes 0-15, 1: lanes 16-31
    const int l16  = lane & 15;

    for (int ti = 0; ti < 9; ++ti) {
        // A (16x4): lanes 0-15 hold M=0..15; VGPR0=K0, VGPR1=K1; lanes 16-31 hold K2,K3.
        v2f a;
        a.x = (half == 0) ? s_mean[ti * 16 + l16] : 0.0f;   // K=0 column = m_i
        a.y = 0.0f;
        for (int tj = 0; tj < 9; ++tj) {
            // B (4x16): VGPR0 lanes 0-15 = row K=0 (N=0..15); all other rows zero.
            v2f bb;
            bb.x = (half == 0) ? s_mean[tj * 16 + l16] : 0.0f;  // row K=0 = m_j
            bb.y = 0.0f;
            v8f c = {};
            c = __builtin_amdgcn_wmma_f32_16x16x4_f32(
                    false, a, false, bb, (short)0, c, false, false);
            // D layout: VGPR r, lanes 0-15 -> M=r, N=lane; lanes 16-31 -> M=r+8, N=lane-16.
#pragma unroll
            for (int r = 0; r < 8; ++r) {
                int i = ti * 16 + r + 8 * half;
                int j = tj * 16 + l16;
                float mi = s_mean[i];
                float mj = s_mean[j];
                float vv = s_valid[i] * s_valid[j];
                float d2 = mi * mi + mj * mj - 2.0f * c[r];  // (m_i - m_j)^2
                acc += vv * __expf(-d2);
            }
        }
    }
    for (int m = 16; m >= 1; m >>= 1) acc += __shfl_xor(acc, m, 32);

    if (lane == 0) {
        float nt        = ntag_acc;
        float pull_loss = (nt > 0.0f) ? pull_acc / fmaxf(nt, 1.0f) : 0.0f;
        float push      = acc - nt;                       // remove diagonal exp(0)=1 terms
        float denom     = fmaxf((nt - 1.0f) * nt, 1.0f);
        float push_loss = (nt > 1.0f) ? push / denom * 0.5f : 0.0f;
        out[b]     = push_loss;   // PUSH_FACTOR = 1.0
        out[B + b] = pull_loss;   // PULL_FACTOR = 1.0
    }
}

extern "C" void kernel_launch(void* const* d_in, const int* in_sizes, int n_in,
                              void* d_out, int out_size, void* d_ws, size_t ws_size,
                              hipStream_t stream) {
    const float* tags = (const float*)d_in[0];
    const int*   gt   = (const int*)d_in[1];
    float*       out  = (float*)d_out;
    float*       ws   = (float*)d_ws;

    const int B = in_sizes[0] / ROW_N;          // 128
    const int ws_elems = B * 3 * K_PAD;         // ~55K floats

    zero_ws_kernel<<<(ws_elems + 255) / 256, 256, 0, stream>>>(ws, ws_elems);

    dim3 grid(CHUNKS, B);
    seg_reduce_kernel<<<grid, BDIM, 0, stream>>>(tags, gt, ws);

    finalize_kernel<<<B, 32, 0, stream>>>(ws, out, B);
}